// KnowledgeEmbedding_36670430773519
// MI455X (gfx1250) — compile-verified
//
#include <hip/hip_runtime.h>

#define B_TOT   65536
#define S_TOT   512
#define D_DIM   100
#define M_TILE  128          // batch rows per workgroup (8 waves x 16)
#define NEG_CHUNK 128        // negatives resident in LDS at a time
#define KSTEPS  (D_DIM / 4)  // 25 WMMA K-steps of 4

typedef float v2f __attribute__((ext_vector_type(2)));
typedef float v8f __attribute__((ext_vector_type(8)));

__device__ __forceinline__ float softplus_f(float x) {
    // log(1+exp(x)) = max(x,0) + log(1+exp(-|x|))  (stable)
    float e = __expf(-fabsf(x));
    return fmaxf(x, 0.0f) + __logf(1.0f + e);
}

__global__ void ke_zero_kernel(float* out) {
    if (threadIdx.x == 0) out[0] = 0.0f;
}

__global__ __launch_bounds__(256)
void ke_loss_kernel(const float* __restrict__ head_table,
                    const float* __restrict__ tail_table,
                    const float* __restrict__ rel_vec,
                    const float* __restrict__ rel_bias,
                    const int*   __restrict__ head_idx,
                    const int*   __restrict__ tail_idx,
                    const int*   __restrict__ neg_idx,
                    float*       __restrict__ out) {
    __shared__ __align__(16) float shNeg[NEG_CHUNK * D_DIM];  // 51200 B
    __shared__ __align__(16) float shBias[M_TILE];            //   512 B

    const int tid  = threadIdx.x;
    const int lane = tid & 31;
    const int wave = tid >> 5;
    const int l15  = lane & 15;
    const int half = lane >> 4;                 // 0 or 1 (K-pair select)
    const int m0   = blockIdx.x * M_TILE;       // block row base
    const int mw   = m0 + wave * 16;            // this wave's row base

    // Per-row relation bias for this block's 128 rows -> LDS
    if (tid < M_TILE) {
        shBias[tid] = rel_bias[tail_idx[m0 + tid]];
    }

    // ---- A fragments: ex = head_table[head_idx] + rel_vec, kept in VGPRs ----
    // fp32 WMMA 16x16x4 A layout: lane l holds row (l&15), K pair
    // {4*kk + 2*(l>>4), 4*kk + 2*(l>>4) + 1} for step kk.
    v2f a[KSTEPS];
    {
        const int hrow = head_idx[mw + l15];
        const float* hp = head_table + (size_t)hrow * D_DIM + 2 * half;
        const float* rp = rel_vec + 2 * half;
#pragma unroll
        for (int kk = 0; kk < KSTEPS; ++kk) {
            float2 h = *(const float2*)(hp + kk * 4);
            float2 r = *(const float2*)(rp + kk * 4);
            a[kk].x = h.x + r.x;
            a[kk].y = h.y + r.y;
        }
    }

    float lane_sum = 0.0f;

    // ---- positive term: softplus(-(dot(t, ex) + bias)) for the wave's 16 rows
    {
        const int trow = tail_idx[mw + l15];
        const float* tp = tail_table + (size_t)trow * D_DIM + 2 * half;
        float p = 0.0f;
#pragma unroll
        for (int kk = 0; kk < KSTEPS; ++kk) {
            float2 t = *(const float2*)(tp + kk * 4);
            p = fmaf(a[kk].x, t.x, p);
            p = fmaf(a[kk].y, t.y, p);
        }
        p += __shfl_xor(p, 16, 32);             // combine the two K-halves
        float pos_logit = p + rel_bias[trow];
        if (lane < 16) lane_sum += softplus_f(-pos_logit);
    }

    __syncthreads();                            // shBias ready

    // Bias registers for this wave's 16 output rows: acc[v] maps to
    // M = v + 8*(lane>>4)  (fp32 C/D layout), N = lane&15.
    float bv[8];
#pragma unroll
    for (int v = 0; v < 8; ++v) bv[v] = shBias[wave * 16 + 8 * half + v];

    // ---- negative logits via V_WMMA_F32_16X16X4_F32, chunked over S ----
    for (int c = 0; c < S_TOT / NEG_CHUNK; ++c) {
        __syncthreads();                        // previous chunk consumed
        // Gather NEG_CHUNK rows of tail_table into LDS.
        // Thread t: row t>>1, half (t&1) of 50 floats (25x float2).
        {
            const int r  = tid >> 1;
            const int h2 = tid & 1;
            const int grow = neg_idx[c * NEG_CHUNK + r];
            const float* gp = tail_table + (size_t)grow * D_DIM + h2 * 50;
            float* sp = shNeg + r * D_DIM + h2 * 50;
#pragma unroll
            for (int j = 0; j < 25; ++j) {
                *(float2*)(sp + 2 * j) = *(const float2*)(gp + 2 * j);
            }
        }
        __syncthreads();

        for (int nt = 0; nt < NEG_CHUNK / 16; ++nt) {
            // fp32 WMMA B layout: lane l holds column (l&15), same K pair
            // selection as A.  neg_logits = ex @ neg.T  ->  B[k][n] = neg[n][k].
            const float* bp = shNeg + (nt * 16 + l15) * D_DIM + 2 * half;
            v8f acc = {};
#pragma unroll
            for (int kk = 0; kk < KSTEPS; ++kk) {
                v2f b = *(const v2f*)(bp + kk * 4);
                acc = __builtin_amdgcn_wmma_f32_16x16x4_f32(
                        false, a[kk], false, b, (short)0, acc, false, false);
            }
#pragma unroll
            for (int v = 0; v < 8; ++v) {
                lane_sum += softplus_f(acc[v] + bv[v]);
            }
        }
    }

    // ---- reduce: wave32 shuffle tree, then one atomic per wave ----
#pragma unroll
    for (int off = 1; off < 32; off <<= 1)
        lane_sum += __shfl_xor(lane_sum, off, 32);
    if (lane == 0)
        atomicAdd(out, lane_sum * (1.0f / (float)B_TOT));
}

extern "C" void kernel_launch(void* const* d_in, const int* in_sizes, int n_in,
                              void* d_out, int out_size, void* d_ws, size_t ws_size,
                              hipStream_t stream) {
    (void)in_sizes; (void)n_in; (void)out_size; (void)d_ws; (void)ws_size;
    const float* head_table = (const float*)d_in[0];
    const float* tail_table = (const float*)d_in[1];
    const float* rel_vec    = (const float*)d_in[2];
    const float* rel_bias   = (const float*)d_in[3];
    const int*   head_idx   = (const int*)d_in[4];
    const int*   tail_idx   = (const int*)d_in[5];
    const int*   neg_idx    = (const int*)d_in[6];
    float* out = (float*)d_out;

    ke_zero_kernel<<<1, 32, 0, stream>>>(out);
    ke_loss_kernel<<<B_TOT / M_TILE, 256, 0, stream>>>(
        head_table, tail_table, rel_vec, rel_bias,
        head_idx, tail_idx, neg_idx, out);
}